// SAMomentumCenter_45406394254010
// MI455X (gfx1250) — compile-verified
//
#include <hip/hip_runtime.h>
#include <hip/hip_bf16.h>
#include <math.h>

#define BATCH 4096
#define DIM   512
#define NCLS  5994
#define NTCOL ((NCLS + 15) / 16)       // 375 column tiles of 16
#define NITER ((NTCOL + 15) / 16)      // 24 iterations of 16 tiles (2 per wave)

typedef float v2f __attribute__((ext_vector_type(2)));
typedef float v8f __attribute__((ext_vector_type(8)));

// ---------------------------------------------------------------------------
// Kernel 1: faithful sequential per-label running average (parallel per class)
// ---------------------------------------------------------------------------
__global__ __launch_bounds__(512) void cadd_kernel(
    const float* __restrict__ x, const int* __restrict__ label,
    float* __restrict__ cadd, float* __restrict__ label_add)
{
  __shared__ int lab[BATCH];
  const int cls = blockIdx.x;
  const int t   = threadIdx.x;
  for (int i = t; i < BATCH; i += 512) lab[i] = label[i];
  __syncthreads();

  float acc = 0.0f;
  int seen = 0;
  for (int i = 0; i < BATCH; ++i) {
    if (lab[i] == cls) {                       // block-uniform branch (LDS broadcast)
      const float xi = x[(size_t)i * DIM + t];
      acc = seen ? 0.5f * (acc + xi) : (acc + xi);
      seen = 1;
    }
  }
  cadd[(size_t)cls * DIM + t] = acc;
  if (t == 0) label_add[cls] = seen ? 1.0f : 0.0f;
}

// ---------------------------------------------------------------------------
// Kernel 2: alpha = sigmoid(centers @ fc_w + b); new_center = gated blend
// ---------------------------------------------------------------------------
__global__ __launch_bounds__(256) void newcenter_kernel(
    const float* __restrict__ centers, const float* __restrict__ fc_w,
    const float* __restrict__ fc_b, const float* __restrict__ cadd,
    const float* __restrict__ label_add, float* __restrict__ newc)
{
  __shared__ float red[256];
  __shared__ float g0s, g1s;
  const int cls = blockIdx.x;
  const int t   = threadIdx.x;
  const float* crow = centers + (size_t)cls * DIM;

  red[t] = crow[t] * fc_w[t] + crow[t + 256] * fc_w[t + 256];
  __syncthreads();
  for (int off = 128; off > 0; off >>= 1) {
    if (t < off) red[t] += red[t + off];
    __syncthreads();
  }
  if (t == 0) {
    const float z     = red[0] + fc_b[0];
    const float alpha = 1.0f / (1.0f + __expf(-z));
    const float seen  = label_add[cls];
    g0s = alpha * seen + (1.0f - seen);
    g1s = (1.0f - alpha) * seen;
  }
  __syncthreads();
  const float g0 = g0s, g1 = g1s;
  const float* carow = cadd + (size_t)cls * DIM;
  float* nrow = newc + (size_t)cls * DIM;
  nrow[t]       = crow[t]       * g0 + carow[t]       * g1;
  nrow[t + 256] = crow[t + 256] * g0 + carow[t + 256] * g1;
}

// ---------------------------------------------------------------------------
// One-shot 16-lane merges (done ONCE per kernel, not per tile)
// ---------------------------------------------------------------------------
__device__ __forceinline__ float bfly_sum16(float v) {
  v += __shfl_xor(v, 8);
  v += __shfl_xor(v, 4);
  v += __shfl_xor(v, 2);
  v += __shfl_xor(v, 1);
  return v;
}
__device__ __forceinline__ void merge_ms16(float& m, float& s) {
#pragma unroll
  for (int off = 8; off >= 1; off >>= 1) {
    const float mo = __shfl_xor(m, off);
    const float so = __shfl_xor(s, off);
    const float M  = fmaxf(m, mo);
    s = s * __expf(m - M) + so * __expf(mo - M);
    m = M;
  }
}
__device__ __forceinline__ void merge_argmax16(float& bv, int& bi) {
#pragma unroll
  for (int off = 8; off >= 1; off >>= 1) {
    const float vo = __shfl_xor(bv, off);
    const int   io = __shfl_xor(bi, off);
    if (vo > bv || (vo == bv && io < bi)) { bv = vo; bi = io; }
  }
}

// ---------------------------------------------------------------------------
// Kernel 3: logits = new_center @ ch_w.T + ch_b; diag CE -> lossC_rows.
// Block = 16 rows; wave w owns column-tile pair {16j+2w, 16j+2w+1}:
// one A fragment feeds two independent WMMA accumulation chains.
// Softmax state is lane-private; merged across 16 lanes once at the end.
// ---------------------------------------------------------------------------
__global__ __launch_bounds__(256) void center_ce_kernel(
    const float* __restrict__ newc, const float* __restrict__ chw,
    const float* __restrict__ chb, float* __restrict__ lossC_rows)
{
  __shared__ float Alds[16][DIM + 4];          // +4 pad: conflict-free b64 frags
  __shared__ float sm_m[8][16], sm_s[8][16], sm_d[8][16];

  const int m0  = blockIdx.x * 16;
  const int tid = threadIdx.x;

  for (int idx = tid; idx < 16 * (DIM / 4); idx += 256) {
    const int row = idx / (DIM / 4);
    const int c4  = idx % (DIM / 4);
    int gr = m0 + row; if (gr >= NCLS) gr = NCLS - 1;
    const float4 v = ((const float4*)(newc + (size_t)gr * DIM))[c4];
    Alds[row][c4 * 4 + 0] = v.x; Alds[row][c4 * 4 + 1] = v.y;
    Alds[row][c4 * 4 + 2] = v.z; Alds[row][c4 * 4 + 3] = v.w;
  }
  __syncthreads();

  const int wave = tid >> 5, lane = tid & 31;
  const int cl = lane & 15, hl = lane >> 4;    // col-in-tile, K-half

  float m[8], s[8], dval[8];
#pragma unroll
  for (int g = 0; g < 8; ++g) { m[g] = -3.0e38f; s[g] = 0.0f; dval[g] = 0.0f; }

  for (int j = 0; j < NITER; ++j) {
    const int nt0 = 16 * j + 2 * wave;
    const int cb0 = nt0 * 16 + cl;
    const int cb1 = cb0 + 16;
    const bool ok0 = (cb0 < NCLS), ok1 = (cb1 < NCLS);
    const int cc0 = ok0 ? cb0 : NCLS - 1;
    const int cc1 = ok1 ? cb1 : NCLS - 1;
    const float* brow0 = chw + (size_t)cc0 * DIM + 2 * hl;
    const float* brow1 = chw + (size_t)cc1 * DIM + 2 * hl;
    const float bias0 = chb[cc0];
    const float bias1 = chb[cc1];

    v8f c0 = {}, c1 = {};
#pragma unroll 4
    for (int k0 = 0; k0 < DIM; k0 += 4) {
      const float2 av  = *(const float2*)&Alds[cl][k0 + 2 * hl];
      const float2 b0v = *(const float2*)(brow0 + k0);
      const float2 b1v = *(const float2*)(brow1 + k0);
      v2f a;  a.x  = av.x;  a.y  = av.y;
      v2f b0; b0.x = b0v.x; b0.y = b0v.y;
      v2f b1; b1.x = b1v.x; b1.y = b1v.y;
      c0 = __builtin_amdgcn_wmma_f32_16x16x4_f32(false, a, false, b0,
                                                 (short)0, c0, false, false);
      c1 = __builtin_amdgcn_wmma_f32_16x16x4_f32(false, a, false, b1,
                                                 (short)0, c1, false, false);
    }
#pragma unroll
    for (int g = 0; g < 8; ++g) {
      const int grow = m0 + g + 8 * hl;
      const float va = ok0 ? (c0[g] + bias0) : -3.0e38f;
      const float vb = ok1 ? (c1[g] + bias1) : -3.0e38f;
      dval[g] += (ok0 && cb0 == grow) ? va : 0.0f;
      dval[g] += (ok1 && cb1 == grow) ? vb : 0.0f;
      const float mn = fmaxf(m[g], fmaxf(va, vb));
      s[g] = s[g] * __expf(m[g] - mn) + __expf(va - mn) + __expf(vb - mn);
      m[g] = mn;
    }
  }
#pragma unroll
  for (int g = 0; g < 8; ++g) {
    merge_ms16(m[g], s[g]);
    dval[g] = bfly_sum16(dval[g]);
  }

  if (cl == 0) {
#pragma unroll
    for (int g = 0; g < 8; ++g) {
      const int r = g + 8 * hl;
      sm_m[wave][r] = m[g]; sm_s[wave][r] = s[g]; sm_d[wave][r] = dval[g];
    }
  }
  __syncthreads();
  if (tid < 16) {
    float M = -3.0e38f;
    for (int w = 0; w < 8; ++w) M = fmaxf(M, sm_m[w][tid]);
    float S = 0.0f, Dg = 0.0f;
    for (int w = 0; w < 8; ++w) {
      S  += sm_s[w][tid] * __expf(sm_m[w][tid] - M);
      Dg += sm_d[w][tid];
    }
    const int gr = m0 + tid;
    if (gr < NCLS) lossC_rows[gr] = (M + __logf(S)) - Dg;
  }
}

// ---------------------------------------------------------------------------
// Kernel 4: logits = x @ ch_w.T + ch_b; label CE + argmax.
// Same dual-tile WMMA structure, lane-private online state.
// ---------------------------------------------------------------------------
__global__ __launch_bounds__(256) void output_ce_kernel(
    const float* __restrict__ x, const int* __restrict__ label,
    const float* __restrict__ chw, const float* __restrict__ chb,
    float* __restrict__ nloss_rows, float* __restrict__ correct_rows)
{
  __shared__ float Alds[16][DIM + 4];
  __shared__ float sm_m[8][16], sm_s[8][16], sm_t[8][16], sm_bm[8][16];
  __shared__ int   sm_bi[8][16];

  const int m0  = blockIdx.x * 16;
  const int tid = threadIdx.x;

  for (int idx = tid; idx < 16 * (DIM / 4); idx += 256) {
    const int row = idx / (DIM / 4);
    const int c4  = idx % (DIM / 4);
    const float4 v = ((const float4*)(x + (size_t)(m0 + row) * DIM))[c4];
    Alds[row][c4 * 4 + 0] = v.x; Alds[row][c4 * 4 + 1] = v.y;
    Alds[row][c4 * 4 + 2] = v.z; Alds[row][c4 * 4 + 3] = v.w;
  }
  __syncthreads();

  const int wave = tid >> 5, lane = tid & 31;
  const int cl = lane & 15, hl = lane >> 4;

  int larr[8];
#pragma unroll
  for (int g = 0; g < 8; ++g) larr[g] = label[m0 + g + 8 * hl];

  float m[8], s[8], tval[8], bm[8];
  int bi[8];
#pragma unroll
  for (int g = 0; g < 8; ++g) {
    m[g] = -3.0e38f; s[g] = 0.0f; tval[g] = 0.0f;
    bm[g] = -3.0e38f; bi[g] = 0x7fffffff;
  }

  for (int j = 0; j < NITER; ++j) {
    const int nt0 = 16 * j + 2 * wave;
    const int cb0 = nt0 * 16 + cl;
    const int cb1 = cb0 + 16;
    const bool ok0 = (cb0 < NCLS), ok1 = (cb1 < NCLS);
    const int cc0 = ok0 ? cb0 : NCLS - 1;
    const int cc1 = ok1 ? cb1 : NCLS - 1;
    const float* brow0 = chw + (size_t)cc0 * DIM + 2 * hl;
    const float* brow1 = chw + (size_t)cc1 * DIM + 2 * hl;
    const float bias0 = chb[cc0];
    const float bias1 = chb[cc1];

    v8f c0 = {}, c1 = {};
#pragma unroll 4
    for (int k0 = 0; k0 < DIM; k0 += 4) {
      const float2 av  = *(const float2*)&Alds[cl][k0 + 2 * hl];
      const float2 b0v = *(const float2*)(brow0 + k0);
      const float2 b1v = *(const float2*)(brow1 + k0);
      v2f a;  a.x  = av.x;  a.y  = av.y;
      v2f b0; b0.x = b0v.x; b0.y = b0v.y;
      v2f b1; b1.x = b1v.x; b1.y = b1v.y;
      c0 = __builtin_amdgcn_wmma_f32_16x16x4_f32(false, a, false, b0,
                                                 (short)0, c0, false, false);
      c1 = __builtin_amdgcn_wmma_f32_16x16x4_f32(false, a, false, b1,
                                                 (short)0, c1, false, false);
    }
#pragma unroll
    for (int g = 0; g < 8; ++g) {
      const float va = ok0 ? (c0[g] + bias0) : -3.0e38f;
      const float vb = ok1 ? (c1[g] + bias1) : -3.0e38f;
      tval[g] += (ok0 && cb0 == larr[g]) ? va : 0.0f;
      tval[g] += (ok1 && cb1 == larr[g]) ? vb : 0.0f;
      // lane-private argmax; columns ascend (cb0 < cb1, j ascending) so
      // strict '>' keeps the first occurrence within this lane.
      if (va > bm[g]) { bm[g] = va; bi[g] = cb0; }
      if (vb > bm[g]) { bm[g] = vb; bi[g] = cb1; }
      const float mn = fmaxf(m[g], fmaxf(va, vb));
      s[g] = s[g] * __expf(m[g] - mn) + __expf(va - mn) + __expf(vb - mn);
      m[g] = mn;
    }
  }
#pragma unroll
  for (int g = 0; g < 8; ++g) {
    merge_ms16(m[g], s[g]);
    tval[g] = bfly_sum16(tval[g]);
    merge_argmax16(bm[g], bi[g]);
  }

  if (cl == 0) {
#pragma unroll
    for (int g = 0; g < 8; ++g) {
      const int r = g + 8 * hl;
      sm_m[wave][r] = m[g]; sm_s[wave][r] = s[g]; sm_t[wave][r] = tval[g];
      sm_bm[wave][r] = bm[g]; sm_bi[wave][r] = bi[g];
    }
  }
  __syncthreads();
  if (tid < 16) {
    float M = -3.0e38f;
    for (int w = 0; w < 8; ++w) M = fmaxf(M, sm_m[w][tid]);
    float S = 0.0f, T = 0.0f, BM = -3.0e38f;
    int BI = 0x7fffffff;
    for (int w = 0; w < 8; ++w) {
      S += sm_s[w][tid] * __expf(sm_m[w][tid] - M);
      T += sm_t[w][tid];
      const float wv = sm_bm[w][tid];
      const int   wi = sm_bi[w][tid];
      if (wv > BM)       { BM = wv; BI = wi; }
      else if (wv == BM) { BI = min(BI, wi); }
    }
    const int gr = m0 + tid;
    nloss_rows[gr]   = (M + __logf(S)) - T;
    correct_rows[gr] = (BI == label[gr]) ? 1.0f : 0.0f;
  }
}

// ---------------------------------------------------------------------------
// Kernel 5: deterministic final reduction -> [loss, prec1]
// ---------------------------------------------------------------------------
__global__ __launch_bounds__(256) void finalize_kernel(
    const float* __restrict__ lossC_rows, const float* __restrict__ nloss_rows,
    const float* __restrict__ correct_rows, float* __restrict__ out)
{
  __shared__ float red0[256], red1[256], red2[256];
  const int t = threadIdx.x;
  float a = 0.0f, b = 0.0f, c = 0.0f;
  for (int i = t; i < NCLS;  i += 256) a += lossC_rows[i];
  for (int i = t; i < BATCH; i += 256) { b += nloss_rows[i]; c += correct_rows[i]; }
  red0[t] = a; red1[t] = b; red2[t] = c;
  __syncthreads();
  for (int off = 128; off > 0; off >>= 1) {
    if (t < off) { red0[t] += red0[t + off]; red1[t] += red1[t + off]; red2[t] += red2[t + off]; }
    __syncthreads();
  }
  if (t == 0) {
    const float lossC = red0[0] / (float)NCLS;
    const float nloss = red1[0] / (float)BATCH;
    out[0] = nloss + lossC;
    out[1] = 100.0f * red2[0] / (float)BATCH;
  }
}

// ---------------------------------------------------------------------------
extern "C" void kernel_launch(void* const* d_in, const int* in_sizes, int n_in,
                              void* d_out, int out_size, void* d_ws, size_t ws_size,
                              hipStream_t stream) {
  const float* x       = (const float*)d_in[0];
  const int*   label   = (const int*)  d_in[1];
  const float* centers = (const float*)d_in[2];
  const float* fc_w    = (const float*)d_in[3];
  const float* fc_b    = (const float*)d_in[4];
  const float* ch_w    = (const float*)d_in[5];
  const float* ch_b    = (const float*)d_in[6];
  float* out = (float*)d_out;

  float* ws         = (float*)d_ws;
  float* cadd       = ws;                              // NCLS*DIM
  float* label_add  = cadd + (size_t)NCLS * DIM;       // 6016 (padded)
  float* newc       = label_add + 6016;                // NCLS*DIM
  float* lossC_rows = newc + (size_t)NCLS * DIM;       // 6016 (padded)
  float* nloss_rows = lossC_rows + 6016;               // BATCH
  float* correct    = nloss_rows + BATCH;              // BATCH

  cadd_kernel<<<NCLS, 512, 0, stream>>>(x, label, cadd, label_add);
  newcenter_kernel<<<NCLS, 256, 0, stream>>>(centers, fc_w, fc_b, cadd, label_add, newc);
  center_ce_kernel<<<NTCOL, 256, 0, stream>>>(newc, ch_w, ch_b, lossC_rows);
  output_ce_kernel<<<BATCH / 16, 256, 0, stream>>>(x, label, ch_w, ch_b, nloss_rows, correct);
  finalize_kernel<<<1, 256, 0, stream>>>(lossC_rows, nloss_rows, correct, out);
}